// MTCA_CapsNet_42391327212082
// MI455X (gfx1250) — compile-verified
//
#include <hip/hip_runtime.h>
#include <hip/hip_bf16.h>
#include <math.h>

typedef __attribute__((ext_vector_type(2))) float v2f;
typedef __attribute__((ext_vector_type(8))) float v8f;

#define B_       2
#define C_       32
#define T_       7680
#define WP       1918      /* conv1 width after stride-2 + avgpool-2 */
#define W2OUT    1910      /* conv2 output width */
#define IN_CAPS  122240
#define OUT_CAPS 16
#define HPERB    977920    /* 16*32*1910 elements per batch */

__device__ __forceinline__ float eluf(float x){ return x > 0.f ? x : (expf(x) - 1.f); }
__device__ __forceinline__ float sigm(float x){ return 1.f / (1.f + expf(-x)); }

// ---------------- K0: zero the small accumulator region -------------------
__global__ void k0_zero(float* ws){
    int i = blockIdx.x * 256 + threadIdx.x;
    if (i < 4096) ws[i] = 0.f;
}

// ---------------- K1a: per-(b,c) max & mean over T ------------------------
__global__ void k1_rowstats(const float* __restrict__ x, float* mx, float* av){
    __shared__ float smax[256], ssum[256];
    int row = blockIdx.x;                       // b*32 + c
    const float* xr = x + (size_t)row * T_;
    float m = -3.402823466e38f, s = 0.f;
    for (int t = threadIdx.x; t < T_; t += 256){ float v = xr[t]; m = fmaxf(m, v); s += v; }
    smax[threadIdx.x] = m; ssum[threadIdx.x] = s; __syncthreads();
    for (int off = 128; off > 0; off >>= 1){
        if (threadIdx.x < off){
            smax[threadIdx.x] = fmaxf(smax[threadIdx.x], smax[threadIdx.x + off]);
            ssum[threadIdx.x] += ssum[threadIdx.x + off];
        }
        __syncthreads();
    }
    if (threadIdx.x == 0){ mx[row] = smax[0]; av[row] = ssum[0] / (float)T_; }
}

// ---------------- K1b: attn = sigmoid((mx+av) @ W_lin^T + 2*b_lin) --------
__global__ void k1b_attn(const float* mx, const float* av, const float* Wlin,
                         const float* blin, float* attn){
    __shared__ float sm[64], sa[64];
    int t = threadIdx.x;
    sm[t] = mx[t]; sa[t] = av[t];
    __syncthreads();
    int b = t >> 5, c = t & 31;
    float acc = 2.f * blin[c];
    for (int c2 = 0; c2 < 32; c2++)
        acc += (sm[b*32 + c2] + sa[b*32 + c2]) * Wlin[c*32 + c2];
    attn[t] = sigm(acc);
}

// ---------------- K2: conv1 (stride 2) + bias + ELU + avgpool2 ------------
__global__ void k2_conv1(const float* __restrict__ x, const float* attn,
                         const float* Wc1, const float* bc1, float* h1){
    int idx = blockIdx.x * 256 + threadIdx.x;
    if (idx >= B_*16*32*WP) return;
    int w  = idx % WP;  int r = idx / WP;
    int hh = r % 32;    r /= 32;
    int oc = r % 16;    int b = r / 16;
    const float* xp = x + (size_t)(b*32 + hh) * T_ + 4*w;
    float a = attn[b*32 + hh];
    float d0 = 0.f, d1 = 0.f;
#pragma unroll
    for (int k = 0; k < 9; k++){ float wv = Wc1[oc*9 + k]; d0 += wv * xp[k]; d1 += wv * xp[k + 2]; }
    float s0 = a*d0 + bc1[oc], s1 = a*d1 + bc1[oc];
    h1[idx] = 0.5f * (eluf(s0) + eluf(s1));
}

// ---------------- K3: conv2 as WMMA GEMM (M=16 oc, K=144, N=w tiles) ------
// blockDim = 64 (2 waves); grid = (60, 32 hh, 2 b); each wave: one 16-col tile
__global__ void k3_conv2_wmma(const float* __restrict__ h1, const float* __restrict__ W2,
                              const float* bc2, float* __restrict__ h2){
    __shared__ float lw[2304];                          // 16oc x 144(ic*9+k)
    for (int i = threadIdx.x; i < 2304; i += 64) lw[i] = W2[i];
    __syncthreads();
    int wave = threadIdx.x >> 5, lane = threadIdx.x & 31;
    int c = lane & 15, hi = lane >> 4;
    int hh = blockIdx.y, b = blockIdx.z;
    int w0 = (blockIdx.x * 2 + wave) * 16;
    v8f acc = {};
    for (int s = 0; s < 36; s++){
        int k0 = 4*s + 2*hi;                            // this lane's K rows: k0, k0+1
        v2f a; a.x = lw[c*144 + k0]; a.y = lw[c*144 + k0 + 1];   // A[oc=c, k]
        int ic0 = k0 / 9,     kk0 = k0 - 9*ic0;
        int ic1 = (k0+1) / 9, kk1 = (k0+1) - 9*ic1;
        int wc0 = w0 + c + kk0; wc0 = wc0 > WP-1 ? WP-1 : wc0;   // clamp tail (not stored)
        int wc1 = w0 + c + kk1; wc1 = wc1 > WP-1 ? WP-1 : wc1;
        v2f bv;
        bv.x = h1[((b*16 + ic0)*32 + hh)*WP + wc0];     // B[k, col=c]
        bv.y = h1[((b*16 + ic1)*32 + hh)*WP + wc1];
        acc = __builtin_amdgcn_wmma_f32_16x16x4_f32(false, a, false, bv,
                                                    (short)0, acc, false, false);
    }
    int w = w0 + c;
#pragma unroll
    for (int r = 0; r < 8; r++){
        int oc = r + 8*hi;
        if (w < W2OUT) h2[((b*16 + oc)*32 + hh)*W2OUT + w] = acc[r] + bc2[oc];
    }
}

// ---------------- K4: squash over capsule dim 8 ---------------------------
__global__ void k4_squash(const float* __restrict__ h2, float* __restrict__ u){
    int idx = blockIdx.x * 256 + threadIdx.x;           // b*IN_CAPS + n
    if (idx >= B_ * IN_CAPS) return;
    const float* s = h2 + (size_t)idx * 8;
    float t[8], nn = 0.f;
#pragma unroll
    for (int d = 0; d < 8; d++){ t[d] = s[d]; nn += t[d]*t[d]; }
    float norm = sqrtf(nn);
    float sc = nn / ((1.f + nn) * (norm + 1e-8f));
    float* o = u + (size_t)idx * 8;
#pragma unroll
    for (int d = 0; d < 8; d++) o[d] = sc * t[d];
}

// ---------------- K5: x_hat via WMMA + fused uniform-c reduction ----------
// grid = (60, 16 j); blockDim = 256 (8 waves); wave handles n stride 480.
// Per n: D(16p x 16) = Wcaps[j,n] (16x8) * B(8x16), B cols 0..1 = u[b,n,:].
__global__ void k5_xhat_wmma(const float* __restrict__ Wcaps, const float* __restrict__ u,
                             float* __restrict__ xhat, float* S0){
    int j = blockIdx.y;
    int wave = threadIdx.x >> 5, lane = threadIdx.x & 31;
    int c = lane & 15, hi = lane >> 4;
    bool isb = (c < 2);
    v8f acc = {};
    for (int n = blockIdx.x * 8 + wave; n < IN_CAPS; n += 480){
        int wb = ((j*IN_CAPS + n)*16 + c)*8 + 2*hi;     // A[p=c, d], two k-steps
        v2f a0, a1;
        a0.x = Wcaps[wb];     a0.y = Wcaps[wb + 1];
        a1.x = Wcaps[wb + 4]; a1.y = Wcaps[wb + 5];
        v2f b0 = {}, b1 = {};
        if (isb){
            int ub = c*HPERB + n*8 + 2*hi;              // B[d, col=b] (cols>=2 zero)
            b0.x = u[ub];     b0.y = u[ub + 1];
            b1.x = u[ub + 4]; b1.y = u[ub + 5];
        }
        v8f d = {};
        d = __builtin_amdgcn_wmma_f32_16x16x4_f32(false, a0, false, b0, (short)0, d, false, false);
        d = __builtin_amdgcn_wmma_f32_16x16x4_f32(false, a1, false, b1, (short)0, d, false, false);
        if (isb){
            int ob = ((c*OUT_CAPS + j)*IN_CAPS + n)*16 + hi*8;
#pragma unroll
            for (int r = 0; r < 8; r++) xhat[ob + r] = d[r];
            acc += d;                                   // fused Sum_n for routing iter 0
        }
        if (wb + 480*128 + 8 < OUT_CAPS*IN_CAPS*128)
            __builtin_prefetch(&Wcaps[wb + 480*128], 0, 1);
    }
    if (isb){
        int sb = (c*OUT_CAPS + j)*16 + hi*8;
        for (int r = 0; r < 8; r++) atomicAdd(&S0[sb + r], acc[r]);
    }
}

// ---------------- K6: v = squash(S*scale); optionally vsum = v + vprev ----
__global__ void k6_squash_v(const float* S, float scale, const float* vprev,
                            float* vout, float* vsum){
    int lane = threadIdx.x;                              // b*16 + j
    float s[16], nn = 0.f;
#pragma unroll
    for (int p = 0; p < 16; p++){ s[p] = S[lane*16 + p] * scale; nn += s[p]*s[p]; }
    float norm = sqrtf(nn);
    float sc = nn / ((1.f + nn) * (norm + 1e-8f));
#pragma unroll
    for (int p = 0; p < 16; p++){
        float v = sc * s[p];
        vout[lane*16 + p] = v;
        if (vsum) vsum[lane*16 + p] = v + vprev[lane*16 + p];
    }
}

// ---------------- K7: routing pass: logits=v.xhat, softmax_j, accum c*xhat
// grid = 120; blockDim = 256 (8 waves); lane = b*16 + j; wave strides over n.
__global__ void k7_route(const float* __restrict__ xhat, const float* __restrict__ vin,
                         float* Sout){
    __shared__ float red[32][16];
    for (int i = threadIdx.x; i < 512; i += 256) ((float*)red)[i] = 0.f;
    __syncthreads();
    int wave = threadIdx.x >> 5, lane = threadIdx.x & 31;
    float vloc[16], accp[16];
#pragma unroll
    for (int p = 0; p < 16; p++){ vloc[p] = vin[lane*16 + p]; accp[p] = 0.f; }
    for (int n = blockIdx.x * 8 + wave; n < IN_CAPS; n += 960){
        const float* xp = xhat + ((size_t)lane * IN_CAPS + n) * 16;
        float xv[16], e = 0.f;
#pragma unroll
        for (int p = 0; p < 16; p++){ xv[p] = xp[p]; e += vloc[p] * xv[p]; }
        float m = e;                                     // softmax over 16 j-lanes
        m = fmaxf(m, __shfl_xor(m, 1)); m = fmaxf(m, __shfl_xor(m, 2));
        m = fmaxf(m, __shfl_xor(m, 4)); m = fmaxf(m, __shfl_xor(m, 8));
        float ex = expf(e - m), ss = ex;
        ss += __shfl_xor(ss, 1); ss += __shfl_xor(ss, 2);
        ss += __shfl_xor(ss, 4); ss += __shfl_xor(ss, 8);
        float cc = ex / ss;
#pragma unroll
        for (int p = 0; p < 16; p++) accp[p] += cc * xv[p];
    }
#pragma unroll
    for (int p = 0; p < 16; p++) atomicAdd(&red[lane][p], accp[p]);  // ds_add_f32
    __syncthreads();
    if (threadIdx.x < 32)
        for (int p = 0; p < 16; p++) atomicAdd(&Sout[threadIdx.x*16 + p], red[threadIdx.x][p]);
}

// ---------------- K10: v2 = squash(S2); out = sigmoid(sum v2*W_fc + b) ----
__global__ void k10_final(const float* S2, const float* Wfc, const float* bfc, float* out){
    int lane = threadIdx.x;
    int j = lane & 15, b = lane >> 4;
    float s[16], nn = 0.f;
#pragma unroll
    for (int p = 0; p < 16; p++){ s[p] = S2[lane*16 + p]; nn += s[p]*s[p]; }
    float norm = sqrtf(nn);
    float sc = nn / ((1.f + nn) * (norm + 1e-8f));
    float part = 0.f;
#pragma unroll
    for (int p = 0; p < 16; p++) part += sc * s[p] * Wfc[j*16 + p];
    part += __shfl_xor(part, 1); part += __shfl_xor(part, 2);
    part += __shfl_xor(part, 4); part += __shfl_xor(part, 8);
    if (j == 0) out[b] = sigm(part + bfc[0]);
}

extern "C" void kernel_launch(void* const* d_in, const int* in_sizes, int n_in,
                              void* d_out, int out_size, void* d_ws, size_t ws_size,
                              hipStream_t stream){
    (void)in_sizes; (void)n_in; (void)out_size; (void)ws_size;
    const float* x     = (const float*)d_in[0];
    const float* Wlin  = (const float*)d_in[1];
    const float* blin  = (const float*)d_in[2];
    const float* Wc1   = (const float*)d_in[3];
    const float* bc1   = (const float*)d_in[4];
    const float* Wc2   = (const float*)d_in[5];
    const float* bc2   = (const float*)d_in[6];
    const float* Wcaps = (const float*)d_in[7];
    const float* Wfc   = (const float*)d_in[8];
    const float* bfc   = (const float*)d_in[9];
    float* out = (float*)d_out;
    float* ws  = (float*)d_ws;

    // workspace layout (floats); total ~68.5M floats (~274 MB)
    float* mx = ws, *av = ws + 64, *attn = ws + 128;
    float* S0 = ws + 192, *S1 = ws + 704, *S2 = ws + 1216;
    float* v0 = ws + 1728, *v1 = ws + 2240, *vs = ws + 2752;
    float* h1 = ws + 4096;
    float* h2 = h1 + (size_t)B_*16*32*WP;
    float* u  = h2 + (size_t)B_*HPERB;
    float* xh = u  + (size_t)B_*HPERB;

    k0_zero      <<<16, 256, 0, stream>>>(ws);
    k1_rowstats  <<<64, 256, 0, stream>>>(x, mx, av);
    k1b_attn     <<<1, 64, 0, stream>>>(mx, av, Wlin, blin, attn);
    int tot1 = B_*16*32*WP;
    k2_conv1     <<<(tot1 + 255)/256, 256, 0, stream>>>(x, attn, Wc1, bc1, h1);
    k3_conv2_wmma<<<dim3(60, 32, 2), 64, 0, stream>>>(h1, Wc2, bc2, h2);
    k4_squash    <<<(B_*IN_CAPS + 255)/256, 256, 0, stream>>>(h2, u);
    k5_xhat_wmma <<<dim3(60, 16), 256, 0, stream>>>(Wcaps, u, xh, S0);
    k6_squash_v  <<<1, 32, 0, stream>>>(S0, 1.f/16.f, nullptr, v0, nullptr);
    k7_route     <<<120, 256, 0, stream>>>(xh, v0, S1);
    k6_squash_v  <<<1, 32, 0, stream>>>(S1, 1.f, v0, v1, vs);   // also vs = v0+v1
    k7_route     <<<120, 256, 0, stream>>>(xh, vs, S2);
    k10_final    <<<1, 32, 0, stream>>>(S2, Wfc, bfc, out);
}